// SceneNet_798863917647
// MI455X (gfx1250) — compile-verified
//
#include <hip/hip_runtime.h>
#include <hip/hip_bf16.h>

// ---------------- problem constants ----------------
#define Wg   96
#define Hg   96
#define NN   (Wg * Hg)        // 9216
#define Dd   132
#define DP   128
#define Bb   2
#define KNB  26
#define MP   9232             // 577 * 16  (rows padded, includes zero row N)
#define KP   144              // 132 padded to 9*16 (multiple of 4 and 16)
#define MT   577
#define NT   9
#define SCALE 0.08703882797784892f   // 1/sqrt(132)

typedef float v2f __attribute__((ext_vector_type(2)));
typedef float v4f __attribute__((ext_vector_type(4)));
typedef float v8f __attribute__((ext_vector_type(8)));

__device__ __forceinline__ int reflect96(int x) {
    // np.pad mode='reflect' with pad=2: -1->1, -2->2, 96->94, 97->93
    return x < 0 ? -x : (x > 95 ? 190 - x : x);
}

// ---------------- 0: zero fill ----------------
__global__ __launch_bounds__(256) void sn_zero(float* __restrict__ p, long long n) {
    long long i = (long long)blockIdx.x * 256 + threadIdx.x;
    if (i < n) p[i] = 0.0f;
}

// ---------------- 0b: pad+transpose weights, pad biases ----------------
// WT[which][n][k] = W[k*132+n] for k,n < 132 else 0   (which: 0=K, 1=Q)
// bp[which][n]    = bias[n] for n < 132 else 0
__global__ __launch_bounds__(256) void sn_prepw(const float* __restrict__ Wk,
                                                const float* __restrict__ bk,
                                                const float* __restrict__ Wq,
                                                const float* __restrict__ bq,
                                                float* __restrict__ wtp,
                                                float* __restrict__ bp) {
    int i = blockIdx.x * 256 + threadIdx.x;
    const int WTN = 2 * KP * KP;           // 41472
    if (i < WTN) {
        int which = i / (KP * KP);
        int r     = i % (KP * KP);
        int n = r / KP, k = r % KP;
        const float* Wm = which ? Wq : Wk;
        wtp[i] = (n < Dd && k < Dd) ? Wm[k * Dd + n] : 0.0f;
    } else if (i < WTN + 2 * KP) {
        int j = i - WTN;
        int which = j / KP, n = j % KP;
        const float* bias = which ? bq : bk;
        bp[j] = (n < Dd) ? bias[n] : 0.0f;
    }
}

// ---------------- 1: L2-normalize features over H axis, scatter to f_pad ----
// features: [B, D, W, H] (H contiguous).  One wave per (b,d,w).
// f_pad: [B, MP, KP], f_pad[b][w*96+h][d] = feat / max(norm,1e-12)
__global__ __launch_bounds__(32) void sn_norm(const float* __restrict__ feat,
                                              float* __restrict__ fpad) {
    int id = blockIdx.x;                  // b*D*W + d*W + w
    int w  = id % Wg;
    int bd = id / Wg;
    int d  = bd % Dd;
    int b  = bd / Dd;
    const float* row = feat + (((size_t)(b * Dd + d)) * Wg + w) * Hg;
    int lane = threadIdx.x;
    float v0 = row[lane], v1 = row[lane + 32], v2 = row[lane + 64];
    float ss = v0 * v0 + v1 * v1 + v2 * v2;
    #pragma unroll
    for (int off = 16; off; off >>= 1) ss += __shfl_xor(ss, off, 32);
    float rinv = 1.0f / fmaxf(sqrtf(ss), 1e-12f);
    size_t base = (((size_t)b * MP) + (size_t)w * Hg) * KP + d;
    fpad[base + (size_t)lane * KP]        = v0 * rinv;
    fpad[base + (size_t)(lane + 32) * KP] = v1 * rinv;
    fpad[base + (size_t)(lane + 64) * KP] = v2 * rinv;
}

// ---------------- 2: GEMM via V_WMMA_F32_16X16X4_F32 ----------------
// ksf/qsf[b] = f_pad[b] @ W + bias, all stored padded [B, MP, KP].
// B-matrix read from pre-transposed padded weights -> branch-free b64 loads.
// grid = (MT, NT, B*2); block = 32 (one wave per 16x16 tile).
__global__ __launch_bounds__(32) void sn_gemm(const float* __restrict__ fpad,
                                              const float* __restrict__ wtp,
                                              const float* __restrict__ bp,
                                              float* __restrict__ ksf,
                                              float* __restrict__ qsf) {
    int mt = blockIdx.x, nt = blockIdx.y, z = blockIdx.z;
    int b = z >> 1;
    int which = z & 1;
    float* out = which ? qsf : ksf;

    int lane  = threadIdx.x;
    int m     = mt * 16 + (lane & 15);        // A row for this lane
    int n     = nt * 16 + (lane & 15);        // output column for this lane
    int khalf = (lane >> 4) << 1;             // 0 for lanes 0-15, 2 for 16-31

    const float* Arow = fpad + ((size_t)b * MP + m) * KP + khalf;
    const float* Brow = wtp + ((size_t)which * KP + n) * KP + khalf;  // WT[n][k]

    v8f c = {};
    #pragma unroll 4
    for (int k0 = 0; k0 < KP; k0 += 4) {
        v2f a  = *(const v2f*)(Arow + k0);
        v2f bf = *(const v2f*)(Brow + k0);
#if defined(__HIP_DEVICE_COMPILE__)
#  if __has_builtin(__builtin_amdgcn_wmma_f32_16x16x4_f32)
        c = __builtin_amdgcn_wmma_f32_16x16x4_f32(false, a, false, bf,
                                                  (short)0, c, false, false);
#  else
#    error "f32 WMMA builtin not available in gfx1250 device pass"
#  endif
#else
        // Host pass only parses device code; keep it well-formed (never runs).
        c[0] += a.x * bf.x + a.y * bf.y;
#endif
    }
    float bv = bp[which * KP + n];
    int rbase = mt * 16 + ((lane >> 4) << 3);   // lanes 0-15: M=r, 16-31: M=8+r
    #pragma unroll
    for (int r = 0; r < 8; ++r) {
        out[((size_t)b * MP + (rbase + r)) * KP + n] = c[r] + bv;
    }
}

// ---------------- 3: edge logits + adjacency weights ----------------
// One wave per (b,n).  lane = neighbor k (0..25 active).
// adjw = exp(logit - rowmax)  ( == softmax(adj)/max(adj) after cancellation )
__global__ __launch_bounds__(32) void sn_logits(const float* __restrict__ ksf,
                                                const float* __restrict__ qsf,
                                                const int* __restrict__ vrand,
                                                float* __restrict__ logits,
                                                float* __restrict__ adjw) {
    int n = blockIdx.x, b = blockIdx.y;
    int lane = threadIdx.x;
    int w0 = n / Hg, h0 = n % Hg;
    int v;
    if (lane < 25) {
        int i = lane / 5, j = lane % 5;
        v = reflect96(w0 + i - 2) * Hg + reflect96(h0 + j - 2);
    } else {
        v = vrand[(size_t)b * NN + n];
    }
    // Rows are 576-byte aligned -> float4 loads are legal.  132 = 33 * 4.
    const v4f* kr = (const v4f*)(ksf + ((size_t)b * MP + n) * KP);
    const v4f* qr = (const v4f*)(qsf + ((size_t)b * MP + v) * KP);
    float acc = 0.0f;
    #pragma unroll 3
    for (int d = 0; d < Dd / 4; ++d) {
        v4f x = kr[d], y = qr[d];
        acc = fmaf(x.x, y.x, acc);
        acc = fmaf(x.y, y.y, acc);
        acc = fmaf(x.z, y.z, acc);
        acc = fmaf(x.w, y.w, acc);
    }
    float l = acc * SCALE;
    bool active = lane < KNB;
    float lm = active ? l : -3.0e38f;
    #pragma unroll
    for (int off = 16; off; off >>= 1) lm = fmaxf(lm, __shfl_xor(lm, off, 32));
    float wgt = __expf(l - lm);
    if (active) {
        size_t o = ((size_t)b * NN + n) * KNB + lane;
        logits[o] = l;
        adjw[o]   = wgt;
    }
}

// ---------------- 4: softmax of h0 over DP=128 ----------------
__global__ __launch_bounds__(32) void sn_hsoft(const float* __restrict__ h0,
                                               float* __restrict__ h) {
    size_t row = blockIdx.x;              // b*N + n
    const float* src = h0 + row * DP;
    float*       dst = h  + row * DP;
    int lane = threadIdx.x;
    float x0 = src[lane], x1 = src[lane + 32], x2 = src[lane + 64], x3 = src[lane + 96];
    float m = fmaxf(fmaxf(x0, x1), fmaxf(x2, x3));
    #pragma unroll
    for (int off = 16; off; off >>= 1) m = fmaxf(m, __shfl_xor(m, off, 32));
    float e0 = __expf(x0 - m), e1 = __expf(x1 - m), e2 = __expf(x2 - m), e3 = __expf(x3 - m);
    float s = e0 + e1 + e2 + e3;
    #pragma unroll
    for (int off = 16; off; off >>= 1) s += __shfl_xor(s, off, 32);
    float r = 1.0f / s;
    dst[lane] = e0 * r; dst[lane + 32] = e1 * r; dst[lane + 64] = e2 * r; dst[lane + 96] = e3 * r;
}

// ---------------- 5: transposed sparse propagation (atomic scatter) --------
// 128 threads per (b,n); thread = channel d.  26 fp32 hardware atomics each.
__global__ __launch_bounds__(128) void sn_scatter(const float* __restrict__ h,
                                                  const float* __restrict__ adjw,
                                                  const int* __restrict__ vrand,
                                                  float* __restrict__ prop) {
    int n = blockIdx.x, b = blockIdx.y;
    int d = threadIdx.x;
    float hv = h[((size_t)b * NN + n) * DP + d];
    const float* wrow = adjw + ((size_t)b * NN + n) * KNB;
    int w0 = n / Hg, h0c = n % Hg;
    #pragma unroll
    for (int k = 0; k < KNB; ++k) {
        int v;
        if (k < 25) {
            int i = k / 5, j = k % 5;
            v = reflect96(w0 + i - 2) * Hg + reflect96(h0c + j - 2);
        } else {
            v = vrand[(size_t)b * NN + n];
        }
        float contrib = wrow[k] * hv;
        __hip_atomic_fetch_add(&prop[((size_t)b * NN + v) * DP + d], contrib,
                               __ATOMIC_RELAXED, __HIP_MEMORY_SCOPE_AGENT);
    }
}

// ---------------- host-side launcher ----------------
extern "C" void kernel_launch(void* const* d_in, const int* in_sizes, int n_in,
                              void* d_out, int out_size, void* d_ws, size_t ws_size,
                              hipStream_t stream) {
    const float* features = (const float*)d_in[0];   // [B, D, W, H]
    const float* Wk       = (const float*)d_in[1];   // [D, D]
    const float* bk       = (const float*)d_in[2];   // [D]
    const float* Wq       = (const float*)d_in[3];   // [D, D]
    const float* bq       = (const float*)d_in[4];   // [D]
    const int*   v_rand   = (const int*)d_in[5];     // [B, N, 1]
    const float* h0       = (const float*)d_in[6];   // [B, N, DP]

    float* out    = (float*)d_out;
    float* logits = out;                               // [B, N, 26]
    float* prop   = out + (size_t)Bb * NN * KNB;       // [B, N, 128]

    float* ws   = (float*)d_ws;
    float* fpad = ws;                                  // B*MP*KP
    float* ksf  = fpad + (size_t)Bb * MP * KP;         // B*MP*KP
    float* qsf  = ksf  + (size_t)Bb * MP * KP;         // B*MP*KP
    float* adjw = qsf  + (size_t)Bb * MP * KP;         // B*N*26
    float* hbuf = adjw + (size_t)Bb * NN * KNB;        // B*N*128
    float* wtp  = hbuf + (size_t)Bb * NN * DP;         // 2*KP*KP (transposed padded W)
    float* bp   = wtp  + (size_t)2 * KP * KP;          // 2*KP    (padded biases)

    // 0: zero padded f and the atomic-accumulated prop output region
    {
        long long nf = (long long)Bb * MP * KP;
        sn_zero<<<dim3((unsigned)((nf + 255) / 256)), dim3(256), 0, stream>>>(fpad, nf);
        long long np = (long long)Bb * NN * DP;
        sn_zero<<<dim3((unsigned)((np + 255) / 256)), dim3(256), 0, stream>>>(prop, np);
    }
    // 0b: pad + transpose weights, pad biases
    {
        int tot = 2 * KP * KP + 2 * KP;
        sn_prepw<<<dim3((tot + 255) / 256), dim3(256), 0, stream>>>(Wk, bk, Wq, bq, wtp, bp);
    }
    // 1: per-(b,d,w) L2 normalization over H + layout transform
    sn_norm<<<dim3(Bb * Dd * Wg), dim3(32), 0, stream>>>(features, fpad);
    // 2: WMMA GEMMs for ksf and qsf (grid.z = batch*2)
    sn_gemm<<<dim3(MT, NT, Bb * 2), dim3(32), 0, stream>>>(fpad, wtp, bp, ksf, qsf);
    // 3: gather dot-products -> logits + adjacency weights
    sn_logits<<<dim3(NN, Bb), dim3(32), 0, stream>>>(ksf, qsf, v_rand, logits, adjw);
    // 4: softmax(h0) over channels
    sn_hsoft<<<dim3(Bb * NN), dim3(32), 0, stream>>>(h0, hbuf);
    // 5: atomic scatter propagation
    sn_scatter<<<dim3(NN, Bb), dim3(128), 0, stream>>>(hbuf, adjw, v_rand, prop);
}